// ACMultiHeadAttention_66709432042332
// MI455X (gfx1250) — compile-verified
//
#include <hip/hip_runtime.h>
#include <hip/hip_bf16.h>

// ---------------- problem constants ----------------
#define BB   32
#define LL   200
#define LP   208        // L padded to 13*16
#define HH   256
#define NHH  4
#define DD   64
#define CC   16         // (2+F)^2
#define TLT  13         // LP/16
#define SCALE_F 0.125f  // 1/sqrt(64)

static constexpr size_t PB = (size_t)BB * LP * HH;        // one padded projection buffer
static constexpr size_t WTS = (size_t)HH * HH;            // one transposed weight (64K floats)
static constexpr size_t SC = (size_t)NHH * LP * CC * LP;  // per-batch score / E1 buffer
static constexpr int    M1 = NHH * LP * CC;               // 13312 rows of the energy GEMM

typedef float v2f __attribute__((ext_vector_type(2)));
typedef float v8f __attribute__((ext_vector_type(8)));

__device__ inline v8f wmma4(v2f a, v2f b, v8f c) {
  return __builtin_amdgcn_wmma_f32_16x16x4_f32(false, a, false, b, (short)0, c, false, false);
}

// ---------------- weight transpose: Wt[n*256+k] = W[k*256+n] ----------------
__global__ __launch_bounds__(256)
void transpose256_kernel(const float* __restrict__ W, float* __restrict__ Wt) {
  int i = blockIdx.x * 256 + threadIdx.x;   // 65536 elements
  int k = i >> 8, n = i & 255;
  Wt[(size_t)n * HH + k] = W[(size_t)k * HH + n];
}

// ------------- pad+transpose Wf1 -> Wf1t[n*LP+k], pad bf1 -> bf1p -------------
__global__ __launch_bounds__(256)
void pad_wf1_kernel(const float* __restrict__ Wf1, const float* __restrict__ bf1,
                    float* __restrict__ Wf1t, float* __restrict__ bf1p) {
  int i = blockIdx.x * blockDim.x + threadIdx.x;
  if (i < LP * LP) {
    int n = i / LP, k = i % LP;
    Wf1t[i] = (k < LL && n < LL) ? Wf1[k * LL + n] : 0.0f;
  }
  if (i < LP) bf1p[i] = (i < LL) ? bf1[i] : 0.0f;
}

// ------- projection GEMM: Out(B,LP,H) = X(B,L,H) @ W + bias, zero pad rows -------
// One wave = 64x16 output (4 M-tiles). B from transposed weight (K-contiguous).
__global__ __launch_bounds__(128)
void proj_gemm_kernel(const float* __restrict__ X, const float* __restrict__ Wt,
                      const float* __restrict__ bias, float* __restrict__ Out) {
  const int ntiles = HH / 16;                      // 16
  const int mgroups = (BB * LP) / 64;              // 104
  int wid = blockIdx.x * (blockDim.x >> 5) + (threadIdx.x >> 5);
  if (wid >= mgroups * ntiles) return;
  const int mg = wid / ntiles, tn = wid % ntiles;
  const int lane = threadIdx.x & 31, half = lane >> 4, ml = lane & 15;
  const int row0 = mg * 64, col0 = tn * 16;

  // branch-free padded-row handling: clamp address, mask value
  const float* Arow[4];
  float amask[4];
  #pragma unroll
  for (int i = 0; i < 4; ++i) {
    const int r = row0 + i * 16 + ml;
    const int bidx = r / LP, l = r % LP;
    amask[i] = (l < LL) ? 1.0f : 0.0f;
    const int lc = (l < LL) ? l : (LL - 1);
    Arow[i] = X + ((size_t)bidx * LL + lc) * HH;
  }
  const float* Brow = Wt + (size_t)(col0 + ml) * HH;

  v8f acc[4] = {{}, {}, {}, {}};
  for (int k0 = 0; k0 < HH; k0 += 4) {
    const int ka = k0 + 2 * half;
    v2f b = *(const v2f*)(Brow + ka);
    #pragma unroll
    for (int i = 0; i < 4; ++i) {
      v2f a = *(const v2f*)(Arow[i] + ka);
      a[0] *= amask[i]; a[1] *= amask[i];
      acc[i] = wmma4(a, b, acc[i]);
    }
  }
  const float bn = bias[col0 + ml];
  #pragma unroll
  for (int i = 0; i < 4; ++i) {
    #pragma unroll
    for (int v = 0; v < 8; ++v) {
      const int rr = row0 + i * 16 + v + 8 * half;
      const int lo = rr % LP;
      Out[(size_t)rr * HH + col0 + ml] = (lo < LL) ? (acc[i][v] + bn) : 0.0f;
    }
  }
}

// ------- scores: per batch b, S[h][l][c][m] = Q_qt(l,:) . K_kt(m,:) -------
// One wave = one (qt,h,tl,tm), all 4 kt tiles (Q fragment reused 4x).
__global__ __launch_bounds__(128)
void scores_kernel(const float* __restrict__ ws, int b, float* __restrict__ Sc) {
  int wid = blockIdx.x * (blockDim.x >> 5) + (threadIdx.x >> 5);
  if (wid >= 4 * NHH * TLT * TLT) return;
  const int qh = wid / (TLT * TLT);
  const int rem = wid % (TLT * TLT);
  const int tl = rem / TLT, tm = rem % TLT;
  const int qt = qh / NHH, h = qh % NHH;
  const int lane = threadIdx.x & 31, half = lane >> 4, ml = lane & 15;

  const float* Qr = ws + (size_t)qt * PB + ((size_t)b * LP + tl * 16 + ml) * HH + h * DD;
  const float* Kr[4];
  #pragma unroll
  for (int kt = 0; kt < 4; ++kt)
    Kr[kt] = ws + (size_t)(4 + kt) * PB + ((size_t)b * LP + tm * 16 + ml) * HH + h * DD;

  v8f acc[4] = {{}, {}, {}, {}};
  for (int k0 = 0; k0 < DD; k0 += 4) {
    const int ka = k0 + 2 * half;
    v2f a = *(const v2f*)(Qr + ka);
    #pragma unroll
    for (int kt = 0; kt < 4; ++kt) {
      v2f bfrag = *(const v2f*)(Kr[kt] + ka);
      acc[kt] = wmma4(a, bfrag, acc[kt]);
    }
  }
  #pragma unroll
  for (int kt = 0; kt < 4; ++kt) {
    const int c = qt * 4 + kt;
    #pragma unroll
    for (int v = 0; v < 8; ++v) {
      const int l = tl * 16 + v + 8 * half;
      const int m = tm * 16 + ml;
      Sc[(((size_t)h * LP + l) * CC + c) * LP + m] = acc[kt][v];
    }
  }
}

// ---------------- E1 = relu(AC @ Wf1 + bf1) : (M1 x LP), 64x16 per wave ----------------
__global__ __launch_bounds__(128)
void e1_kernel(const float* __restrict__ Sc, const float* __restrict__ Wf1t,
               const float* __restrict__ bf1p, float* __restrict__ E1) {
  const int mgroups = M1 / 64;   // 208
  int wid = blockIdx.x * (blockDim.x >> 5) + (threadIdx.x >> 5);
  if (wid >= mgroups * TLT) return;
  const int mg = wid / TLT, tn = wid % TLT;
  const int lane = threadIdx.x & 31, half = lane >> 4, ml = lane & 15;
  const int row0 = mg * 64, col0 = tn * 16;

  const float* Arow[4];
  #pragma unroll
  for (int i = 0; i < 4; ++i) Arow[i] = Sc + (size_t)(row0 + i * 16 + ml) * LP;
  const float* Brow = Wf1t + (size_t)(col0 + ml) * LP;

  __builtin_prefetch(Brow, 0, 3);
  __builtin_prefetch(Arow[0], 0, 1);
  __builtin_prefetch(Arow[1], 0, 1);
  __builtin_prefetch(Arow[2], 0, 1);
  __builtin_prefetch(Arow[3], 0, 1);

  v8f acc[4] = {{}, {}, {}, {}};
  for (int k0 = 0; k0 < LP; k0 += 4) {
    const int ka = k0 + 2 * half;
    v2f b = *(const v2f*)(Brow + ka);
    #pragma unroll
    for (int i = 0; i < 4; ++i) {
      v2f a = *(const v2f*)(Arow[i] + ka);
      acc[i] = wmma4(a, b, acc[i]);
    }
  }
  const float bn = bf1p[col0 + ml];
  #pragma unroll
  for (int i = 0; i < 4; ++i) {
    #pragma unroll
    for (int v = 0; v < 8; ++v) {
      const int rr = row0 + i * 16 + v + 8 * half;
      E1[(size_t)rr * LP + col0 + ml] = fmaxf(acc[i][v] + bn, 0.0f);
    }
  }
}

// ---------------- E2: energy[r] = E1[r,:200] . Wf2 + bf2 (one wave per row) ----------------
__global__ __launch_bounds__(128)
void e2_kernel(const float* __restrict__ E1, const float* __restrict__ Wf2,
               const float* __restrict__ bf2, float* __restrict__ energy) {
  int r = blockIdx.x * (blockDim.x >> 5) + (threadIdx.x >> 5);
  if (r >= M1) return;
  const int lane = threadIdx.x & 31;
  const float* row = E1 + (size_t)r * LP;
  float s = 0.0f;
  for (int n = lane; n < LL; n += 32) s += row[n] * Wf2[n];
  for (int off = 16; off > 0; off >>= 1) s += __shfl_down(s, off, 32);
  if (lane == 0) energy[r] = s + bf2[0];
}

// ---------- fused: w=softmax_C(energy); attn=sum_c w*S*SCALE+mask; probs=softmax_m ----------
__global__ __launch_bounds__(256)
void attn_kernel(const float* __restrict__ Sc, const float* __restrict__ energy,
                 const float* __restrict__ mask, float* __restrict__ probs) {
  const int h = blockIdx.x / LP, l = blockIdx.x % LP;
  const int t = threadIdx.x;
  float* prow = probs + ((size_t)h * LP + l) * LP;
  if (l >= LL) { if (t < LP) prow[t] = 0.0f; return; }

  __shared__ float shw[CC];
  __shared__ float sh[256];
  if (t == 0) {
    const float* en = energy + ((size_t)h * LP + l) * CC;
    float mx = en[0];
    for (int c = 1; c < CC; ++c) mx = fmaxf(mx, en[c]);
    float s = 0.0f;
    for (int c = 0; c < CC; ++c) { float e = __expf(en[c] - mx); shw[c] = e; s += e; }
    float inv = 1.0f / s;
    for (int c = 0; c < CC; ++c) shw[c] *= inv;
  }
  __syncthreads();

  float val = -3.0e38f;
  if (t < LL) {
    const float* srow = Sc + (((size_t)h * LP + l) * CC) * LP + t;
    float acc = 0.0f;
    #pragma unroll
    for (int c = 0; c < CC; ++c) acc += shw[c] * srow[(size_t)c * LP];
    val = acc * SCALE_F + mask[l * LL + t];
  }
  sh[t] = val; __syncthreads();
  for (int s = 128; s > 0; s >>= 1) { if (t < s) sh[t] = fmaxf(sh[t], sh[t + s]); __syncthreads(); }
  const float mx = sh[0]; __syncthreads();
  const float e = (t < LL) ? __expf(val - mx) : 0.0f;
  sh[t] = e; __syncthreads();
  for (int s = 128; s > 0; s >>= 1) { if (t < s) sh[t] += sh[t + s]; __syncthreads(); }
  const float inv = 1.0f / sh[0];
  if (t < LP) prow[t] = e * inv;
}

// ---------------- ctx = probs @ V : per batch, per head (208 x 64) ----------------
__global__ __launch_bounds__(128)
void ctx_kernel(const float* __restrict__ probs, const float* __restrict__ ws,
                int b, float* __restrict__ ctx) {
  int wid = blockIdx.x * (blockDim.x >> 5) + (threadIdx.x >> 5);
  if (wid >= NHH * TLT * 4) return;
  const int h = wid / (TLT * 4);
  const int rem = wid % (TLT * 4);
  const int tl = rem / 4, tn = rem % 4;
  const int lane = threadIdx.x & 31, half = lane >> 4, ml = lane & 15;

  const float* Arow = probs + ((size_t)h * LP + tl * 16 + ml) * LP;
  const float* Vb = ws + 8 * PB + ((size_t)b * LP) * HH + h * DD + tn * 16 + ml; // [k*HH]
  v8f acc = {};
  for (int k0 = 0; k0 < LP; k0 += 4) {
    const int ka = k0 + 2 * half;
    v2f a = *(const v2f*)(Arow + ka);
    v2f bfrag;
    bfrag[0] = Vb[(size_t)ka * HH];
    bfrag[1] = Vb[(size_t)(ka + 1) * HH];
    acc = wmma4(a, bfrag, acc);
  }
  #pragma unroll
  for (int v = 0; v < 8; ++v) {
    const int l = tl * 16 + v + 8 * half;
    ctx[((size_t)b * LP + l) * HH + h * DD + tn * 16 + ml] = acc[v];
  }
}

// ---------------- x = ctx @ Wd + bd + input  (padded rows -> 0), 64x16 per wave ----------------
__global__ __launch_bounds__(128)
void final_gemm_kernel(const float* __restrict__ ctx, const float* __restrict__ Wdt,
                       const float* __restrict__ bd, const float* __restrict__ input,
                       float* __restrict__ xbuf) {
  const int ntiles = HH / 16;
  const int mgroups = (BB * LP) / 64;
  int wid = blockIdx.x * (blockDim.x >> 5) + (threadIdx.x >> 5);
  if (wid >= mgroups * ntiles) return;
  const int mg = wid / ntiles, tn = wid % ntiles;
  const int lane = threadIdx.x & 31, half = lane >> 4, ml = lane & 15;
  const int row0 = mg * 64, col0 = tn * 16;

  const float* Arow[4];
  #pragma unroll
  for (int i = 0; i < 4; ++i) Arow[i] = ctx + (size_t)(row0 + i * 16 + ml) * HH;
  const float* Brow = Wdt + (size_t)(col0 + ml) * HH;

  v8f acc[4] = {{}, {}, {}, {}};
  for (int k0 = 0; k0 < HH; k0 += 4) {
    const int ka = k0 + 2 * half;
    v2f b = *(const v2f*)(Brow + ka);
    #pragma unroll
    for (int i = 0; i < 4; ++i) {
      v2f a = *(const v2f*)(Arow[i] + ka);
      acc[i] = wmma4(a, b, acc[i]);
    }
  }
  const int n = col0 + ml;
  const float bn = bd[n];
  #pragma unroll
  for (int i = 0; i < 4; ++i) {
    #pragma unroll
    for (int v = 0; v < 8; ++v) {
      const int rr = row0 + i * 16 + v + 8 * half;
      const int bidx = rr / LP, l = rr % LP;
      const float m = (l < LL) ? 1.0f : 0.0f;
      const int lc = (l < LL) ? l : (LL - 1);
      const float res = input[((size_t)bidx * LL + lc) * HH + n];
      xbuf[(size_t)rr * HH + n] = m * (acc[i][v] + bn + res);
    }
  }
}

// ---------------- LayerNorm over H=256 (one block per (b,l) row) ----------------
__global__ __launch_bounds__(256)
void ln_kernel(const float* __restrict__ xbuf, const float* __restrict__ gamma,
               const float* __restrict__ beta, float* __restrict__ out) {
  const int r = blockIdx.x;
  const int bidx = r / LL, l = r % LL;
  const int t = threadIdx.x;
  const float x = xbuf[((size_t)bidx * LP + l) * HH + t];
  __shared__ float sh[HH];
  sh[t] = x; __syncthreads();
  for (int s = 128; s > 0; s >>= 1) { if (t < s) sh[t] += sh[t + s]; __syncthreads(); }
  const float mu = sh[0] * (1.0f / HH); __syncthreads();
  const float d = x - mu;
  sh[t] = d * d; __syncthreads();
  for (int s = 128; s > 0; s >>= 1) { if (t < s) sh[t] += sh[t + s]; __syncthreads(); }
  const float var = sh[0] * (1.0f / HH);
  out[((size_t)bidx * LL + l) * HH + t] = d * rsqrtf(var + 1e-12f) * gamma[t] + beta[t];
}

// ---------------- host launch ----------------
extern "C" void kernel_launch(void* const* d_in, const int* in_sizes, int n_in,
                              void* d_out, int out_size, void* d_ws, size_t ws_size,
                              hipStream_t stream) {
  const float* input = (const float*)d_in[0];
  const float* attr  = (const float*)d_in[1];   // (F,B,L,1,H)
  const float* pos   = (const float*)d_in[2];
  const float* mask  = (const float*)d_in[3];   // (1,1,L,L)
  const float* Wq  = (const float*)d_in[4];  const float* bq  = (const float*)d_in[5];
  const float* Wk  = (const float*)d_in[6];  const float* bk  = (const float*)d_in[7];
  const float* Wv  = (const float*)d_in[8];  const float* bv  = (const float*)d_in[9];
  const float* Wqp = (const float*)d_in[10]; const float* bqp = (const float*)d_in[11];
  const float* Wkp = (const float*)d_in[12]; const float* bkp = (const float*)d_in[13];
  // d_in[14..15] = Wvp,bvp (unused);  d_in[20..21] = Wv_a,bv_a (unused)
  const float* Wq_a = (const float*)d_in[16]; const float* bq_a = (const float*)d_in[17];
  const float* Wk_a = (const float*)d_in[18]; const float* bk_a = (const float*)d_in[19];
  const float* Wf1 = (const float*)d_in[22]; const float* bf1 = (const float*)d_in[23];
  const float* Wf2 = (const float*)d_in[24]; const float* bf2 = (const float*)d_in[25];
  const float* Wd  = (const float*)d_in[26]; const float* bd  = (const float*)d_in[27];
  const float* gamma = (const float*)d_in[28];
  const float* beta  = (const float*)d_in[29];
  float* out = (float*)d_out;

  float* ws = (float*)d_ws;
  // workspace layout (floats):
  float* P      = ws;                          // 9 padded projection buffers, PB each
  float* WT     = ws + 9 * PB;                 // 10 transposed 256x256 weights
  float* Wf1t   = WT + 10 * WTS;               // LP*LP
  float* bf1p   = Wf1t + (size_t)LP * LP;      // LP
  float* scores = bf1p + LP;                   // SC  (per-batch)
  float* E1     = scores + SC;                 // SC  (per-batch)
  float* energy = E1 + SC;                     // M1
  float* probs  = energy + M1;                 // NH*LP*LP (per-batch)
  float* ctx    = probs + (size_t)NHH * LP * LP; // PB
  float* xbuf   = ctx + PB;                    // PB
  (void)ws_size; (void)n_in; (void)in_sizes; (void)out_size;

  // transposed weights: order matches projection order below + Wd last
  const size_t AW = (size_t)HH * HH;
  const float* Wsrc[10] = {Wq, Wqp, Wq_a, Wq_a + AW, Wk, Wk_a, Wk_a + AW, Wkp, Wv, Wd};
  for (int i = 0; i < 10; ++i)
    transpose256_kernel<<<256, 256, 0, stream>>>(Wsrc[i], WT + (size_t)i * WTS);
  pad_wf1_kernel<<<(LP * LP + 255) / 256, 256, 0, stream>>>(Wf1, bf1, Wf1t, bf1p);

  // 9 projections. Q order: q_i, q_p, qa0, qa1. K order: k_i, ka0, ka1, k_p. Then v_i.
  const int projWaves = ((BB * LP) / 64) * (HH / 16);   // 1664
  const int projBlocks = (projWaves + 3) / 4;
  const size_t ATT = (size_t)BB * LL * HH;
  const float* Xsrc[9] = {input, pos, attr, attr + ATT, input, attr, attr + ATT, pos, input};
  const float* bsrc[9] = {bq, bqp, bq_a, bq_a + HH, bk, bk_a, bk_a + HH, bkp, bv};
  for (int i = 0; i < 9; ++i)
    proj_gemm_kernel<<<projBlocks, 128, 0, stream>>>(Xsrc[i], WT + (size_t)i * WTS,
                                                     bsrc[i], P + (size_t)i * PB);

  const int scoreWaves = 4 * NHH * TLT * TLT;   // 2704 (each wave does 4 kt tiles)
  const int e1Waves    = (M1 / 64) * TLT;       // 2704
  const int ctxWaves   = NHH * TLT * 4;         // 208
  for (int b = 0; b < BB; ++b) {
    scores_kernel<<<(scoreWaves + 3) / 4, 128, 0, stream>>>(ws, b, scores);
    e1_kernel<<<(e1Waves + 3) / 4, 128, 0, stream>>>(scores, Wf1t, bf1p, E1);
    e2_kernel<<<(M1 + 3) / 4, 128, 0, stream>>>(E1, Wf2, bf2, energy);
    attn_kernel<<<NHH * LP, 256, 0, stream>>>(scores, energy, mask, probs);
    ctx_kernel<<<(ctxWaves + 3) / 4, 128, 0, stream>>>(probs, ws, b, ctx);
  }

  final_gemm_kernel<<<projBlocks, 128, 0, stream>>>(ctx, WT + 9 * WTS, bd, input, xbuf);
  ln_kernel<<<BB * LL, 256, 0, stream>>>(xbuf, gamma, beta, out);
}